// DecoderUpBlock_69569880261095
// MI455X (gfx1250) — compile-verified
//
#include <hip/hip_runtime.h>
#include <hip/hip_bf16.h>

// ---------------------------------------------------------------------------
// RetSeg3D DecoderUpBlock for MI455X (gfx1250, wave32).
// fp32 end-to-end with V_WMMA_F32_16X16X4_F32.
// - Weights staged per kernel-offset into LDS via the Tensor Data Mover
//   (tensor_load_to_lds + s_wait_tensorcnt), then pair-interleaved so each
//   B fragment is one ds_load_b64.
// - Two 16-row M-tiles per wave: 8 WMMAs per 4 LDS loads + 2 global loads.
// - Scatter-add with hardware global_atomic_add_f32 (unsafeAtomicAdd).
// ---------------------------------------------------------------------------

typedef __attribute__((ext_vector_type(2))) float v2f;
typedef __attribute__((ext_vector_type(8))) float v8f;
typedef __attribute__((ext_vector_type(4))) unsigned int v4u;
typedef __attribute__((ext_vector_type(8))) int v8i;
typedef __attribute__((ext_vector_type(4))) int v4i;

#if defined(__HIP_DEVICE_COMPILE__) && __has_builtin(__builtin_amdgcn_tensor_load_to_lds)
#define HAVE_TDM 1
#else
#define HAVE_TDM 0
#endif

#define EPS 1e-4f
#define N_IN   60000
#define N_OUT  120000
#define C_IN   128
#define C_OUT  64
#define P_UP   40000
#define Q_SUB  90000
#define K_UP   8
#define K_SUB  27

// ---------------------------------------------------------------------------
// Elementwise BN + ReLU, float4 vectorized. C is 64 or 128 (power of two).
// ---------------------------------------------------------------------------
__global__ __launch_bounds__(256) void bnrelu_kernel(
    const float* __restrict__ X, const float* __restrict__ g,
    const float* __restrict__ b, const float* __restrict__ m,
    const float* __restrict__ v, float* __restrict__ Y, int n4, int C) {
  int i = blockIdx.x * 256 + threadIdx.x;
  if (i >= n4) return;
  float4 x = ((const float4*)X)[i];
  int c = (i << 2) & (C - 1);
  float4 y;
  y.x = fmaxf((x.x - m[c + 0]) * rsqrtf(v[c + 0] + EPS) * g[c + 0] + b[c + 0], 0.f);
  y.y = fmaxf((x.y - m[c + 1]) * rsqrtf(v[c + 1] + EPS) * g[c + 1] + b[c + 1], 0.f);
  y.z = fmaxf((x.z - m[c + 2]) * rsqrtf(v[c + 2] + EPS) * g[c + 2] + b[c + 2], 0.f);
  y.w = fmaxf((x.w - m[c + 3]) * rsqrtf(v[c + 3] + EPS) * g[c + 3] + b[c + 3], 0.f);
  ((float4*)Y)[i] = y;
}

// feat[:, 0:64] = lateral ; feat[:, 64:128] = 0   (row stride 128 floats)
__global__ __launch_bounds__(256) void init_feat_kernel(
    const float* __restrict__ lat, float* __restrict__ feat, int n4) {
  int i = blockIdx.x * 256 + threadIdx.x;  // over N_OUT*32 float4 slots
  if (i >= n4) return;
  int row = i >> 5, q = i & 31;
  float4 z = {0.f, 0.f, 0.f, 0.f};
  ((float4*)feat)[i] = (q < 16) ? ((const float4*)lat)[row * 16 + q] : z;
}

__global__ __launch_bounds__(256) void zero_kernel(float* __restrict__ p, int n4) {
  int i = blockIdx.x * 256 + threadIdx.x;
  if (i >= n4) return;
  float4 z = {0.f, 0.f, 0.f, 0.f};
  ((float4*)p)[i] = z;
}

__global__ __launch_bounds__(256) void copy_kernel(const float* __restrict__ s,
                                                   float* __restrict__ d, int n4) {
  int i = blockIdx.x * 256 + threadIdx.x;
  if (i >= n4) return;
  ((float4*)d)[i] = ((const float4*)s)[i];
}

// ---------------------------------------------------------------------------
// Core: NT (1 or 2) 16-row tiles x 64 cols per wave. B fragments come from
// pair-interleaved LDS: il[(k/2)*128 + 2*c + (k&1)] == W[k][c], so each
// fragment {W[k][c], W[k+1][c]} is one aligned ds_load_b64.
// ---------------------------------------------------------------------------
template <int CIN, int NT>
__device__ __forceinline__ void gms_tiles(
    const float* __restrict__ X, const float* __restrict__ il,
    const int* __restrict__ rinP, const int* __restrict__ routP,
    float* __restrict__ OUT, int ldout, int lane) {
  const int m     = lane & 15;        // column-within-tile / row-within-16
  const int kh    = (lane >> 4) << 1; // K sub-offset: 0 (lanes 0-15) / 2 (16-31)
  const int mBase = (lane >> 4) << 3;

  const float* Arow[NT];
#pragma unroll
  for (int t = 0; t < NT; ++t) {
    const int r = rinP[t * 16 + m];   // lanes L and L+16 read the same index
    Arow[t] = X + (size_t)r * CIN + kh;
  }

  v8f acc[NT][4];
#pragma unroll
  for (int t = 0; t < NT; ++t)
#pragma unroll
    for (int n = 0; n < 4; ++n) acc[t][n] = (v8f){};

  for (int k4 = 0; k4 < CIN; k4 += 4) {
    const float* bp = il + (size_t)(((k4 + kh) >> 1) * 128);
    v2f b0 = *(const v2f*)(bp + 2 * ( 0 + m));
    v2f b1 = *(const v2f*)(bp + 2 * (16 + m));
    v2f b2 = *(const v2f*)(bp + 2 * (32 + m));
    v2f b3 = *(const v2f*)(bp + 2 * (48 + m));
#pragma unroll
    for (int t = 0; t < NT; ++t) {
      v2f a = *(const v2f*)(Arow[t] + k4);
      acc[t][0] = __builtin_amdgcn_wmma_f32_16x16x4_f32(false, a, false, b0, (short)0, acc[t][0], false, false);
      acc[t][1] = __builtin_amdgcn_wmma_f32_16x16x4_f32(false, a, false, b1, (short)0, acc[t][1], false, false);
      acc[t][2] = __builtin_amdgcn_wmma_f32_16x16x4_f32(false, a, false, b2, (short)0, acc[t][2], false, false);
      acc[t][3] = __builtin_amdgcn_wmma_f32_16x16x4_f32(false, a, false, b3, (short)0, acc[t][3], false, false);
    }
  }

  // D layout: acc element i -> row (mBase+i), col = m + 16*n.
#pragma unroll
  for (int t = 0; t < NT; ++t) {
#pragma unroll
    for (int i = 0; i < 8; ++i) {
      const int row = routP[t * 16 + mBase + i];
      float* o = OUT + (size_t)row * ldout + m;
      unsafeAtomicAdd(o +  0, acc[t][0][i]);
      unsafeAtomicAdd(o + 16, acc[t][1][i]);
      unsafeAtomicAdd(o + 32, acc[t][2][i]);
      unsafeAtomicAdd(o + 48, acc[t][3][i]);
    }
  }
}

// ---------------------------------------------------------------------------
// Rulebook gather -> WMMA fp32 GEMM -> atomic scatter-add.
// blockIdx.y = kernel offset k. W[k] (CIN x 64 fp32) is DMA'd into LDS by the
// Tensor Data Mover (wave 0 issues tensor_load_to_lds, waits TENSORcnt), then
// pair-interleaved by the whole block. Dynamic LDS: [raw CIN*64 | il CIN*64].
// ---------------------------------------------------------------------------
template <int CIN>
__global__ __launch_bounds__(256) void gms_wmma_kernel(
    const float* __restrict__ X, const float* __restrict__ W,
    const int* __restrict__ rin, const int* __restrict__ rout,
    float* __restrict__ OUT, int P, int ldout) {
  extern __shared__ float lds[];
  float* raw = lds;             // TDM destination (dynamic-LDS base, offset 0)
  float* il  = lds + CIN * 64;  // pair-interleaved weights

  const int k = blockIdx.y;
  const float* Wk = W + (size_t)k * CIN * 64;

#if HAVE_TDM
  if (threadIdx.x < 32) {
    const unsigned long long ga = (unsigned long long)(uintptr_t)Wk;
    // D# group0: count=1, lds_addr=0, global_addr, type=2 ("image").
    v4u g0 = {1u, 0u, (unsigned)(ga & 0xffffffffu),
              (unsigned)(((ga >> 32) & 0x1ffffffu) | 0x80000000u)};
    // D# group1: data_size=4B; tensor_dim0=64, tensor_dim1=CIN;
    // tile_dim0=64, tile_dim1=CIN; tensor_dim0_stride=64.
    v8i g1 = {(int)(2u << 16),
              (int)(64u << 16),
              (int)(((unsigned)CIN & 0xffffu) << 16),
              (int)(64u << 16),
              (int)CIN,
              64,
              0, 0};
    v4i g2 = {0, 0, 0, 0};
    v4i g3 = {0, 0, 0, 0};
#if __clang_major__ >= 23
    v8i g4 = {0, 0, 0, 0, 0, 0, 0, 0};
    __builtin_amdgcn_tensor_load_to_lds(g0, g1, g2, g3, g4, 0);
#else
    __builtin_amdgcn_tensor_load_to_lds(g0, g1, g2, g3, 0);
#endif
    __builtin_amdgcn_s_wait_tensorcnt(0);
  }
  __syncthreads();
  for (int idx = threadIdx.x; idx < CIN * 64; idx += 256) {
    const int kk = idx >> 6, c = idx & 63;
    il[(((kk >> 1) * 64 + c) << 1) + (kk & 1)] = raw[idx];
  }
#else
  for (int idx = threadIdx.x; idx < CIN * 64; idx += 256) {
    const int kk = idx >> 6, c = idx & 63;
    il[(((kk >> 1) * 64 + c) << 1) + (kk & 1)] = Wk[idx];
  }
#endif
  __syncthreads();

  const int wave = threadIdx.x >> 5;
  const int lane = threadIdx.x & 31;
  const int tiles = P >> 4;
  const int tile0 = (blockIdx.x * 8 + wave) * 2;
  if (tile0 >= tiles) return;

  const int* rinP  = rin  + (size_t)k * P + tile0 * 16;
  const int* routP = rout + (size_t)k * P + tile0 * 16;
  __builtin_prefetch(routP, 0, 0);  // global_prefetch_b8 the scatter indices

  if (tile0 + 1 < tiles)
    gms_tiles<CIN, 2>(X, il, rinP, routP, OUT, ldout, lane);
  else
    gms_tiles<CIN, 1>(X, il, rinP, routP, OUT, ldout, lane);
}

// ---------------------------------------------------------------------------
// Dense residual: OUT[i,:] = ADD[i,:] + X[i,:] @ Wg   (X:[N,128], Wg:[128,64])
// Same WMMA mapping, contiguous rows, two tiles per wave, no atomics.
// ---------------------------------------------------------------------------
__global__ __launch_bounds__(256) void dense_res_wmma_kernel(
    const float* __restrict__ X, const float* __restrict__ Wg,
    const float* __restrict__ ADD, float* __restrict__ OUT, int nTiles) {
  __shared__ float il[128 * 64];
  for (int idx = threadIdx.x; idx < 128 * 64; idx += 256) {
    const int kk = idx >> 6, c = idx & 63;
    il[(((kk >> 1) * 64 + c) << 1) + (kk & 1)] = Wg[idx];
  }
  __syncthreads();

  const int wave = threadIdx.x >> 5;
  const int lane = threadIdx.x & 31;
  const int tile0 = (blockIdx.x * 8 + wave) * 2;
  if (tile0 >= nTiles) return;  // nTiles is even: both tiles valid when any is

  const int m     = lane & 15;
  const int kh    = (lane >> 4) << 1;
  const int mBase = (lane >> 4) << 3;

  const float* Arow[2];
#pragma unroll
  for (int t = 0; t < 2; ++t)
    Arow[t] = X + (size_t)((tile0 + t) * 16 + m) * 128 + kh;

  v8f acc[2][4];
#pragma unroll
  for (int t = 0; t < 2; ++t)
#pragma unroll
    for (int n = 0; n < 4; ++n) acc[t][n] = (v8f){};

  for (int k4 = 0; k4 < 128; k4 += 4) {
    const float* bp = il + (size_t)(((k4 + kh) >> 1) * 128);
    v2f b0 = *(const v2f*)(bp + 2 * ( 0 + m));
    v2f b1 = *(const v2f*)(bp + 2 * (16 + m));
    v2f b2 = *(const v2f*)(bp + 2 * (32 + m));
    v2f b3 = *(const v2f*)(bp + 2 * (48 + m));
#pragma unroll
    for (int t = 0; t < 2; ++t) {
      v2f a = *(const v2f*)(Arow[t] + k4);
      acc[t][0] = __builtin_amdgcn_wmma_f32_16x16x4_f32(false, a, false, b0, (short)0, acc[t][0], false, false);
      acc[t][1] = __builtin_amdgcn_wmma_f32_16x16x4_f32(false, a, false, b1, (short)0, acc[t][1], false, false);
      acc[t][2] = __builtin_amdgcn_wmma_f32_16x16x4_f32(false, a, false, b2, (short)0, acc[t][2], false, false);
      acc[t][3] = __builtin_amdgcn_wmma_f32_16x16x4_f32(false, a, false, b3, (short)0, acc[t][3], false, false);
    }
  }

#pragma unroll
  for (int t = 0; t < 2; ++t) {
#pragma unroll
    for (int i = 0; i < 8; ++i) {
      const int row = (tile0 + t) * 16 + mBase + i;
      const float* ad = ADD + (size_t)row * 64 + m;
      float* o = OUT + (size_t)row * 64 + m;
      o[ 0] = acc[t][0][i] + ad[ 0];
      o[16] = acc[t][1][i] + ad[16];
      o[32] = acc[t][2][i] + ad[32];
      o[48] = acc[t][3][i] + ad[48];
    }
  }
}

// ---------------------------------------------------------------------------
extern "C" void kernel_launch(void* const* d_in, const int* in_sizes, int n_in,
                              void* d_out, int out_size, void* d_ws, size_t ws_size,
                              hipStream_t stream) {
  const float* inp_feats = (const float*)d_in[0];
  const float* inp_up    = (const float*)d_in[1];
  const float* up_g = (const float*)d_in[2];
  const float* up_b = (const float*)d_in[3];
  const float* up_m = (const float*)d_in[4];
  const float* up_v = (const float*)d_in[5];
  const float* W_up = (const float*)d_in[6];
  const float* b0n1_g = (const float*)d_in[7];
  const float* b0n1_b = (const float*)d_in[8];
  const float* b0n1_m = (const float*)d_in[9];
  const float* b0n1_v = (const float*)d_in[10];
  const float* W_b0c1 = (const float*)d_in[11];
  const float* b0n2_g = (const float*)d_in[12];
  const float* b0n2_b = (const float*)d_in[13];
  const float* b0n2_m = (const float*)d_in[14];
  const float* b0n2_v = (const float*)d_in[15];
  const float* W_b0c2 = (const float*)d_in[16];
  const float* W_b0res = (const float*)d_in[17];
  const float* b1n1_g = (const float*)d_in[18];
  const float* b1n1_b = (const float*)d_in[19];
  const float* b1n1_m = (const float*)d_in[20];
  const float* b1n1_v = (const float*)d_in[21];
  const float* W_b1c1 = (const float*)d_in[22];
  const float* b1n2_g = (const float*)d_in[23];
  const float* b1n2_b = (const float*)d_in[24];
  const float* b1n2_m = (const float*)d_in[25];
  const float* b1n2_v = (const float*)d_in[26];
  const float* W_b1c2 = (const float*)d_in[27];
  const int* rb_up_in  = (const int*)d_in[28];
  const int* rb_up_out = (const int*)d_in[29];
  const int* rb_sub_in  = (const int*)d_in[30];
  const int* rb_sub_out = (const int*)d_in[31];
  float* out = (float*)d_out;

  // Workspace partition (floats).
  float* ws  = (float*)d_ws;
  float* B0  = ws;                               // [N_IN , 128] bnrelu(inp_feats)
  float* B1  = B0 + (size_t)N_IN * 128;          // [N_OUT, 128] feat (concat)
  float* B2  = B1 + (size_t)N_OUT * 128;         // [N_OUT, 128] bnrelu(feat)
  float* C0  = B2 + (size_t)N_OUT * 128;         // [N_OUT, 64]
  float* C1  = C0 + (size_t)N_OUT * 64;          // [N_OUT, 64]
  float* C2b = C1 + (size_t)N_OUT * 64;          // [N_OUT, 64]
  float* C3  = C2b + (size_t)N_OUT * 64;         // [N_OUT, 64] feat2

  const int n4_in128  = N_IN  * 128 / 4;
  const int n4_out128 = N_OUT * 128 / 4;
  const int n4_out64  = N_OUT * 64 / 4;
  const int g_in128   = (n4_in128  + 255) / 256;
  const int g_out128  = (n4_out128 + 255) / 256;
  const int g_out64   = (n4_out64  + 255) / 256;

  const int tilesUp  = P_UP / 16;    // 2500
  const int tilesSub = Q_SUB / 16;   // 5625
  const dim3 gridUp(((tilesUp + 1) / 2 + 7) / 8, K_UP);
  const dim3 gridSub(((tilesSub + 1) / 2 + 7) / 8, K_SUB);
  const int denseTiles = N_OUT / 16;                       // 7500 (even)
  const int gDense = ((denseTiles / 2) + 7) / 8;
  const size_t lds128 = 2u * 128 * 64 * sizeof(float);     // 64 KB
  const size_t lds64  = 2u * 64 * 64 * sizeof(float);      // 32 KB

  // 1) x = bnrelu(inp_feats)
  bnrelu_kernel<<<g_in128, 256, 0, stream>>>(inp_feats, up_g, up_b, up_m, up_v,
                                             B0, n4_in128, 128);
  // 2) feat = [lateral | 0]
  init_feat_kernel<<<(N_OUT * 32 + 255) / 256, 256, 0, stream>>>(inp_up, B1, N_OUT * 32);
  // 3) feat[:,64:128] += upconv(x)
  gms_wmma_kernel<128><<<gridUp, 256, lds128, stream>>>(B0, W_up, rb_up_in, rb_up_out,
                                                        B1 + 64, P_UP, 128);
  // 4) y = bnrelu(feat)
  bnrelu_kernel<<<g_out128, 256, 0, stream>>>(B1, b0n1_g, b0n1_b, b0n1_m, b0n1_v,
                                              B2, n4_out128, 128);
  // 5-6) C0 = gms(y, W_b0c1)
  zero_kernel<<<g_out64, 256, 0, stream>>>(C0, n4_out64);
  gms_wmma_kernel<128><<<gridSub, 256, lds128, stream>>>(B2, W_b0c1, rb_sub_in, rb_sub_out,
                                                         C0, Q_SUB, 64);
  // 7) C1 = bnrelu(C0)
  bnrelu_kernel<<<g_out64, 256, 0, stream>>>(C0, b0n2_g, b0n2_b, b0n2_m, b0n2_v,
                                             C1, n4_out64, 64);
  // 8-9) C2b = gms(C1, W_b0c2)
  zero_kernel<<<g_out64, 256, 0, stream>>>(C2b, n4_out64);
  gms_wmma_kernel<64><<<gridSub, 256, lds64, stream>>>(C1, W_b0c2, rb_sub_in, rb_sub_out,
                                                       C2b, Q_SUB, 64);
  // 10) feat2 = C2b + feat @ W_b0res
  dense_res_wmma_kernel<<<gDense, 256, 0, stream>>>(B1, W_b0res, C2b, C3, denseTiles);
  // 11) C0 = bnrelu(feat2)
  bnrelu_kernel<<<g_out64, 256, 0, stream>>>(C3, b1n1_g, b1n1_b, b1n1_m, b1n1_v,
                                             C0, n4_out64, 64);
  // 12-13) C1 = gms(C0, W_b1c1)
  zero_kernel<<<g_out64, 256, 0, stream>>>(C1, n4_out64);
  gms_wmma_kernel<64><<<gridSub, 256, lds64, stream>>>(C0, W_b1c1, rb_sub_in, rb_sub_out,
                                                       C1, Q_SUB, 64);
  // 14) C2b = bnrelu(C1)
  bnrelu_kernel<<<g_out64, 256, 0, stream>>>(C1, b1n2_g, b1n2_b, b1n2_m, b1n2_v,
                                             C2b, n4_out64, 64);
  // 15) out = feat2
  copy_kernel<<<g_out64, 256, 0, stream>>>(C3, out, n4_out64);
  // 16) out += gms(C2b, W_b1c2)
  gms_wmma_kernel<64><<<gridSub, 256, lds64, stream>>>(C2b, W_b1c2, rb_sub_in, rb_sub_out,
                                                       out, Q_SUB, 64);
}